// ResidualGCNLayer_47373489274964
// MI455X (gfx1250) — compile-verified
//
#include <hip/hip_runtime.h>

#define DF 128            // feature dim
#define BN_EPS 1e-5f

typedef float v2f __attribute__((ext_vector_type(2)));
typedef float v8f __attribute__((ext_vector_type(8)));

// ---------------------------------------------------------------- utilities
__global__ void zero_f32_kernel(float* __restrict__ p, int n) {
    int i = blockIdx.x * blockDim.x + threadIdx.x;
    if (i < n) p[i] = 0.0f;
}

// ------------------------------------------------------------- degree count
__global__ void deg_kernel(const long long* __restrict__ dst,
                           float* __restrict__ deg, int e) {
    int i = blockIdx.x * blockDim.x + threadIdx.x;
    if (i < e) atomicAdd(&deg[(int)dst[i]], 1.0f);
}

__global__ void dinv_kernel(const float* __restrict__ deg,
                            float* __restrict__ dinv, int n) {
    int i = blockIdx.x * blockDim.x + threadIdx.x;
    if (i < n) dinv[i] = rsqrtf(deg[i] + 1.0f);   // +1 self loop
}

// ------------------------------------------------------ fp32 WMMA GEMM X*W
// Block = 256 threads = 8 waves. Each wave computes 16 rows x 128 cols.
// W staged in LDS TRANSPOSED with a pair-swizzle:
//   element (k, n) of W lives at  sW[n*128 + 2*((k/2 + 2n) & 63) + (k&1)]
// so each B fragment (W[k0][n], W[k0+1][n]) is one aligned ds_load_b64, and
// the swizzle spreads the 32 lanes across all 64 LDS banks (conflict-free).
// A-row index is CLAMPED (not masked): an out-of-range A row only feeds an
// out-of-range output row, which is never stored -> branchless b64 loads.
__global__ __launch_bounds__(256)
void gemm_wmma_kernel(const float* __restrict__ X,
                      const float* __restrict__ W,
                      float* __restrict__ lin, int n_rows) {
    __shared__ float sW[DF * DF];                 // 64 KB of the 320 KB LDS

    // transpose + pair-swizzle staging (coalesced global reads, one-time)
    for (int i = threadIdx.x; i < DF * DF; i += blockDim.x) {
        const int k = i >> 7;                     // row of W (K index)
        const int n = i & 127;                    // col of W (N index)
        const int q = ((k >> 1) + 2 * n) & 63;    // swizzled pair slot
        sW[n * DF + 2 * q + (k & 1)] = W[i];
    }
    __syncthreads();

    const int wave = threadIdx.x >> 5;            // wave32
    const int lane = threadIdx.x & 31;
    const int m16  = lane & 15;
    const int half = lane >> 4;                   // 0: K+0/1, 1: K+2/3
    const int row0 = (blockIdx.x * 8 + wave) * 16;
    if (row0 >= n_rows) return;                   // wave-uniform exit: EXEC stays all-1

    v8f acc[8] = {};                              // 8 n-tiles of 16x16 f32

    const int arow = min(row0 + m16, n_rows - 1); // clamp, never mask
    const float* __restrict__ arowp = X + (size_t)arow * DF;

    for (int kk = 0; kk < DF; kk += 4) {
        const int k0 = kk + 2 * half;             // ISA A layout: v0=K0/K2, v1=K1/K3
        const v2f a = *(const v2f*)(arowp + k0);  // contiguous, 8B aligned -> b64 load
        const int p0 = k0 >> 1;                   // k-pair index (k0 is even)
#pragma unroll
        for (int t = 0; t < 8; ++t) {
            const int n = t * 16 + m16;           // B column for this lane
            const int q = (p0 + 2 * n) & 63;
            const v2f b = *(const v2f*)(sW + n * DF + 2 * q);  // ds_load_b64
            acc[t] = __builtin_amdgcn_wmma_f32_16x16x4_f32(
                /*neg_a=*/false, a, /*neg_b=*/false, b,
                /*c_mod=*/(short)0, acc[t], /*reuse_a=*/false, /*reuse_b=*/false);
        }
    }

    // C/D layout: VGPR v -> lanes 0-15: M=v, lanes 16-31: M=v+8; N = lane&15
    if (row0 + 16 <= n_rows) {                    // wave-uniform fast path
#pragma unroll
        for (int t = 0; t < 8; ++t) {
            const int n0 = t * 16;
#pragma unroll
            for (int v = 0; v < 8; ++v) {
                const int row = row0 + v + 8 * half;
                lin[(size_t)row * DF + n0 + m16] = acc[t][v];
            }
        }
    } else {                                      // tail block only
#pragma unroll
        for (int t = 0; t < 8; ++t) {
            const int n0 = t * 16;
#pragma unroll
            for (int v = 0; v < 8; ++v) {
                const int row = row0 + v + 8 * half;
                if (row < n_rows) lin[(size_t)row * DF + n0 + m16] = acc[t][v];
            }
        }
    }
}

// --------------------------------------------- self-loop + bias: h init
__global__ void self_bias_kernel(const float* __restrict__ lin,
                                 const float* __restrict__ dinv,
                                 const float* __restrict__ bias,
                                 float* __restrict__ h, int n) {
    int gid = blockIdx.x * blockDim.x + threadIdx.x;
    if (gid >= n * (DF / 4)) return;
    const int row = gid >> 5;                     // 32 float4 chunks per row
    const int c   = (gid & 31) * 4;
    const float di = dinv[row];
    const float w  = di * di;
    const float4 v  = *(const float4*)(lin  + (size_t)row * DF + c);
    const float4 bb = *(const float4*)(bias + c);
    float4 r;
    r.x = v.x * w + bb.x;  r.y = v.y * w + bb.y;
    r.z = v.z * w + bb.z;  r.w = v.w * w + bb.w;
    *(float4*)(h + (size_t)row * DF + c) = r;
}

// -------------------------------------------- edge scatter (L2-resident)
// One wave per edge: 32 lanes x float4 = 128 features.
__global__ void scatter_kernel(const long long* __restrict__ ei,
                               const float* __restrict__ lin,
                               const float* __restrict__ dinv,
                               float* __restrict__ h, int e) {
    int gid = blockIdx.x * blockDim.x + threadIdx.x;
    const int edge = gid >> 5;
    if (edge >= e) return;
    const int c = (gid & 31) * 4;
    const int s = (int)ei[edge];                  // src
    const int d = (int)ei[e + edge];              // dst
    const float norm = dinv[s] * dinv[d];
    const float4 v = *(const float4*)(lin + (size_t)s * DF + c);
    float* p = h + (size_t)d * DF + c;
    atomicAdd(p + 0, v.x * norm);
    atomicAdd(p + 1, v.y * norm);
    atomicAdd(p + 2, v.z * norm);
    atomicAdd(p + 3, v.w * norm);
}

// ------------------------------------------------------- BN column stats
__global__ __launch_bounds__(DF)
void bn_stats_kernel(const float* __restrict__ h,
                     float* __restrict__ stats, int n) {
    const int d = threadIdx.x;                    // column, blockDim = 128
    const int rows_per = (n + gridDim.x - 1) / gridDim.x;
    const int r0 = blockIdx.x * rows_per;
    const int r1 = min(n, r0 + rows_per);
    float s = 0.0f, s2 = 0.0f;
    for (int r = r0; r < r1; ++r) {
        const float v = h[(size_t)r * DF + d];    // coalesced
        s += v; s2 += v * v;
    }
    atomicAdd(&stats[d], s);
    atomicAdd(&stats[DF + d], s2);
}

__global__ __launch_bounds__(DF)
void bn_finalize_kernel(const float* __restrict__ gamma,
                        const float* __restrict__ beta,
                        float* __restrict__ stats, float inv_n) {
    const int d = threadIdx.x;
    const float mean = stats[d] * inv_n;
    const float var  = stats[DF + d] * inv_n - mean * mean;
    const float sc   = gamma[d] * rsqrtf(var + BN_EPS);
    stats[2 * DF + d] = sc;
    stats[3 * DF + d] = beta[d] - mean * sc;
}

// ------------------------------------------- fused BN apply + ReLU + res
__global__ void out_kernel(const float* __restrict__ h,
                           const float* __restrict__ stats,
                           const float* __restrict__ x,
                           float* __restrict__ out, int n) {
    int gid = blockIdx.x * blockDim.x + threadIdx.x;
    if (gid >= n * (DF / 4)) return;
    const int row = gid >> 5;
    const int c   = (gid & 31) * 4;
    const float4 hv = *(const float4*)(h + (size_t)row * DF + c);
    const float4 xv = *(const float4*)(x + (size_t)row * DF + c);
    const float4 sc = *(const float4*)(stats + 2 * DF + c);
    const float4 sh = *(const float4*)(stats + 3 * DF + c);
    float4 r;
    r.x = fmaxf(hv.x * sc.x + sh.x, 0.0f) + xv.x;
    r.y = fmaxf(hv.y * sc.y + sh.y, 0.0f) + xv.y;
    r.z = fmaxf(hv.z * sc.z + sh.z, 0.0f) + xv.z;
    r.w = fmaxf(hv.w * sc.w + sh.w, 0.0f) + xv.w;
    *(float4*)(out + (size_t)row * DF + c) = r;
}

// ---------------------------------------------------------------- launcher
extern "C" void kernel_launch(void* const* d_in, const int* in_sizes, int n_in,
                              void* d_out, int out_size, void* d_ws, size_t ws_size,
                              hipStream_t stream) {
    const float*     x     = (const float*)d_in[0];
    const long long* ei    = (const long long*)d_in[1];   // int64 (2,E)
    const float*     W     = (const float*)d_in[2];
    const float*     bias  = (const float*)d_in[3];
    const float*     gamma = (const float*)d_in[4];
    const float*     beta  = (const float*)d_in[5];
    float* out = (float*)d_out;

    const int n = in_sizes[0] / DF;       // 50000
    const int e = in_sizes[1] / 2;        // 600000
    const size_t nd = (size_t)n * DF;

    // workspace layout (floats)
    float* ws    = (float*)d_ws;
    float* deg   = ws;                    // n
    float* dinv  = ws + n;                // n
    float* lin   = ws + 2 * (size_t)n;    // n*DF
    float* h     = lin + nd;              // n*DF
    float* stats = h + nd;                // 4*DF: sum, sumsq, scale, shift

    const int T = 256;

    // 1) degree + dinv
    zero_f32_kernel<<<(n + T - 1) / T, T, 0, stream>>>(deg, n);
    deg_kernel<<<(e + T - 1) / T, T, 0, stream>>>(ei + e, deg, e);
    dinv_kernel<<<(n + T - 1) / T, T, 0, stream>>>(deg, dinv, n);

    // 2) lin = X * W  (fp32 WMMA, 128 rows per block)
    gemm_wmma_kernel<<<(n + 127) / 128, T, 0, stream>>>(x, W, lin, n);

    // 3) h = lin*dinv^2 + b, then edge scatter
    const int chunks = n * (DF / 4);
    self_bias_kernel<<<(chunks + T - 1) / T, T, 0, stream>>>(lin, dinv, bias, h, n);
    const long long sthreads = (long long)e * 32;
    scatter_kernel<<<(int)((sthreads + T - 1) / T), T, 0, stream>>>(ei, lin, dinv, h, e);

    // 4) batch-norm stats -> scale/shift
    zero_f32_kernel<<<1, 512, 0, stream>>>(stats, 4 * DF);
    bn_stats_kernel<<<512, DF, 0, stream>>>(h, stats, n);
    bn_finalize_kernel<<<1, DF, 0, stream>>>(gamma, beta, stats, 1.0f / (float)n);

    // 5) BN apply + ReLU + residual
    out_kernel<<<(chunks + T - 1) / T, T, 0, stream>>>(h, stats, x, out, n);
}